// SMBLDMesh_38860864094314
// MI455X (gfx1250) — compile-verified
//
#include <hip/hip_runtime.h>
#include <hip/hip_bf16.h>

// SMBLD mesh skinning pipeline for gfx1250 (MI455X).
// Dense contractions run on V_WMMA_F32_16X16X4_F32 (fp32 in/out, matching the
// fp32 reference). Big blend GEMMs use 16x64 register blocking (4 accumulators
// per wave, one A fragment reused 4x) + global_prefetch for the streaming B
// panel. Skinning stages the per-batch joint-transform panel into LDS with the
// Tensor Data Mover (TENSORcnt / s_wait_tensorcnt) and feeds WMMA from DS.

#define NB    512         // batch
#define NV    3889
#define NJ    35
#define V3    11667       // NV*3
#define NBETA 20
#define NPF   306         // pose feature length

typedef float v2f __attribute__((ext_vector_type(2)));
typedef float v8f __attribute__((ext_vector_type(8)));
typedef unsigned int v4u __attribute__((ext_vector_type(4)));
typedef int v4i __attribute__((ext_vector_type(4)));
typedef int v8i __attribute__((ext_vector_type(8)));

__device__ __constant__ int c_parents[NJ] = {
    0, 0, 1, 2, 3, 4, 5, 6, 7, 8, 9, 6, 11, 12, 13, 6, 15, 16, 17, 18,
    16, 20, 21, 22, 23, 16, 25, 26, 27, 28, 29, 30, 16, 1, 1};

// ---------------------------------------------------------------------------
// Generic 16x16 tile: A[m*ldaRow + k*ldaK] (zero-padded past mvalid),
// B[k*ldbK + n] (zero-padded past nvalid). EXEC is full at every WMMA; edge
// guards are value-selects on clamped addresses.
// ---------------------------------------------------------------------------
__device__ __forceinline__ v8f wmma_tile_f32(const float* __restrict__ A,
                                             long ldaRow, long ldaK, int mvalid,
                                             const float* __restrict__ B,
                                             long ldbK, int nvalid, int K) {
  const int lane = threadIdx.x & 31;
  const int rc   = lane & 15;
  const int kb   = (lane >> 4) << 1;     // 0 or 2
  const int arow = (rc < mvalid) ? rc : (mvalid - 1);
  const int bcol = (rc < nvalid) ? rc : (nvalid - 1);
  const float am = (rc < mvalid) ? 1.0f : 0.0f;
  const float bm = (rc < nvalid) ? 1.0f : 0.0f;
  const float* Ap = A + (long)arow * ldaRow;

  v8f c = {0.f, 0.f, 0.f, 0.f, 0.f, 0.f, 0.f, 0.f};
  const int Kmain = K & ~3;
  int k = 0;
  for (; k < Kmain; k += 4) {
    v2f a, b;
    a.x = am * Ap[(long)(k + kb)     * ldaK];
    a.y = am * Ap[(long)(k + kb + 1) * ldaK];
    b.x = bm * B[(long)(k + kb)     * ldbK + bcol];
    b.y = bm * B[(long)(k + kb + 1) * ldbK + bcol];
    c = __builtin_amdgcn_wmma_f32_16x16x4_f32(false, a, false, b,
                                              (short)0, c, false, false);
  }
  if (k < K) {
    v2f a = {0.f, 0.f}, b = {0.f, 0.f};
    if (k + kb < K) {
      a.x = am * Ap[(long)(k + kb) * ldaK];
      b.x = bm * B[(long)(k + kb) * ldbK + bcol];
    }
    if (k + kb + 1 < K) {
      a.y = am * Ap[(long)(k + kb + 1) * ldaK];
      b.y = bm * B[(long)(k + kb + 1) * ldbK + bcol];
    }
    c = __builtin_amdgcn_wmma_f32_16x16x4_f32(false, a, false, b,
                                              (short)0, c, false, false);
  }
  return c;
}

// ---------------------------------------------------------------------------
// Register-blocked 16x64 strip: full 16 rows of A with contiguous K
// (A fragment is one b64 load, reused by 4 WMMAs). B is row-major KxV3;
// streaming rows get prefetched 16 K-steps ahead.
// ---------------------------------------------------------------------------
__device__ __forceinline__ void wmma_strip16x64_f32(const float* __restrict__ A,
                                                    long lda,
                                                    const float* __restrict__ B,
                                                    long ldb, int ncols, int K,
                                                    v8f acc[4]) {
  const int lane = threadIdx.x & 31;
  const int rc = lane & 15;
  const int kb = (lane >> 4) << 1;
  const float* Ap = A + (long)rc * lda;

  int bc[4];
  float bm[4];
  const v8f zero = {0.f, 0.f, 0.f, 0.f, 0.f, 0.f, 0.f, 0.f};
  #pragma unroll
  for (int t = 0; t < 4; ++t) {
    const int cidx = t * 16 + rc;
    bc[t] = (cidx < ncols) ? cidx : 0;
    bm[t] = (cidx < ncols) ? 1.0f : 0.0f;
    acc[t] = zero;
  }

  const int Kmain = K & ~3;
  int k = 0;
  for (; k < Kmain; k += 4) {
    const v2f a = *reinterpret_cast<const v2f*>(Ap + k + kb);  // b64 load
    const float* B0 = B + (long)(k + kb) * ldb;
    const float* B1 = B0 + ldb;
    if (k + 16 < Kmain)  // pull streaming B rows into WGP$ ahead of use
      __builtin_prefetch(B + (long)(k + 16 + kb) * ldb + bc[0], 0, 1);
    #pragma unroll
    for (int t = 0; t < 4; ++t) {
      v2f b;
      b.x = bm[t] * B0[bc[t]];
      b.y = bm[t] * B1[bc[t]];
      acc[t] = __builtin_amdgcn_wmma_f32_16x16x4_f32(false, a, false, b,
                                                     (short)0, acc[t],
                                                     false, false);
    }
  }
  if (k < K) {
    v2f a = {0.f, 0.f};
    if (k + kb < K)     a.x = Ap[k + kb];
    if (k + kb + 1 < K) a.y = Ap[k + kb + 1];
    #pragma unroll
    for (int t = 0; t < 4; ++t) {
      v2f b = {0.f, 0.f};
      if (k + kb < K)     b.x = bm[t] * B[(long)(k + kb) * ldb + bc[t]];
      if (k + kb + 1 < K) b.y = bm[t] * B[(long)(k + kb + 1) * ldb + bc[t]];
      acc[t] = __builtin_amdgcn_wmma_f32_16x16x4_f32(false, a, false, b,
                                                     (short)0, acc[t],
                                                     false, false);
    }
  }
}

// K1: v_shaped = beta(512x20) @ shapedirs(20x11667) + v_template + deform
__global__ void k_shape_blend(const float* __restrict__ beta,
                              const float* __restrict__ shapedirs,
                              const float* __restrict__ v_template,
                              const float* __restrict__ deform,
                              float* __restrict__ v_shaped) {
  const int g = blockIdx.x;              // 64-column group (183 groups)
  const int ty = blockIdx.y;             // 16-row tile over 512
  const int col0 = g * 64;
  const int ncols = (V3 - col0 < 64) ? (V3 - col0) : 64;

  v8f acc[4];
  wmma_strip16x64_f32(beta + (long)ty * 16 * NBETA, NBETA,
                      shapedirs + col0, V3, ncols, NBETA, acc);

  const int lane = threadIdx.x & 31;
  const int cl = lane & 15, rb = (lane >> 4) << 3;
  #pragma unroll
  for (int t = 0; t < 4; ++t) {
    const int cc = t * 16 + cl;
    if (cc < ncols) {
      const int cg = col0 + cc;
      const float vt = v_template[cg];
      #pragma unroll
      for (int i = 0; i < 8; ++i) {
        const int n = ty * 16 + rb + i;
        const long o = (long)n * V3 + cg;
        v_shaped[o] = acc[t][i] + vt + deform[o];
      }
    }
  }
}

// K2/K7: dst[n,j,c] = sum_v src[n,v,c] * Jr[v,j]   (src is (N,V,3) flat)
__global__ void k_jreg(const float* __restrict__ src,
                       const float* __restrict__ Jr,
                       float* __restrict__ dst) {
  const int tj = blockIdx.x;             // col tile over 35 (3 tiles)
  const int tn = blockIdx.y;             // row tile over 512
  const int ch = blockIdx.z;             // channel 0..2
  const int ncols = (NJ - tj * 16 < 16) ? (NJ - tj * 16) : 16;

  v8f acc = wmma_tile_f32(src + (long)tn * 16 * V3 + ch, V3, 3, 16,
                          Jr + (long)tj * 16, NJ, ncols, NV);

  const int lane = threadIdx.x & 31;
  const int col = lane & 15, rb = (lane >> 4) << 3;
  if (col < ncols) {
    const int j = tj * 16 + col;
    #pragma unroll
    for (int i = 0; i < 8; ++i) {
      const int n = tn * 16 + rb + i;
      dst[(long)n * (NJ * 3) + j * 3 + ch] = acc[i];
    }
  }
}

// K3: Rodrigues -> Rs (output) and pose_feature (Rs[1:] - I)
__global__ void k_rodrigues(const float* __restrict__ theta,
                            float* __restrict__ Rs,
                            float* __restrict__ pf) {
  const int idx = blockIdx.x * blockDim.x + threadIdx.x;
  if (idx >= NB * NJ) return;
  const int n = idx / NJ, j = idx % NJ;
  const float* t = theta + (long)n * (NJ * 3) + j * 3;
  const float x = t[0], y = t[1], z = t[2];
  const float ang = sqrtf(x * x + y * y + z * z + 1e-8f);
  const float rx = x / ang, ry = y / ang, rz = z / ang;
  const float c = cosf(ang), s = sinf(ang), ic = 1.0f - c;
  float R[9];
  R[0] = c + ic * rx * rx;       R[1] = ic * rx * ry - s * rz;  R[2] = ic * rx * rz + s * ry;
  R[3] = ic * ry * rx + s * rz;  R[4] = c + ic * ry * ry;       R[5] = ic * ry * rz - s * rx;
  R[6] = ic * rz * rx - s * ry;  R[7] = ic * rz * ry + s * rx;  R[8] = c + ic * rz * rz;
  float* ro = Rs + (long)n * (NJ * 9) + j * 9;
  #pragma unroll
  for (int e = 0; e < 9; ++e) ro[e] = R[e];
  if (j > 0) {
    float* po = pf + (long)n * NPF + (j - 1) * 9;
    #pragma unroll
    for (int e = 0; e < 9; ++e)
      po[e] = R[e] - ((e == 0 || e == 4 || e == 8) ? 1.0f : 0.0f);
  }
}

// K4: v_posed = pose_feature(512x306) @ posedirs(306x11667) + v_shaped (in place)
__global__ void k_pose_blend(const float* __restrict__ pf,
                             const float* __restrict__ posedirs,
                             float* __restrict__ v_sp) {
  const int g = blockIdx.x;
  const int ty = blockIdx.y;
  const int col0 = g * 64;
  const int ncols = (V3 - col0 < 64) ? (V3 - col0) : 64;

  v8f acc[4];
  wmma_strip16x64_f32(pf + (long)ty * 16 * NPF, NPF,
                      posedirs + col0, V3, ncols, NPF, acc);

  const int lane = threadIdx.x & 31;
  const int cl = lane & 15, rb = (lane >> 4) << 3;
  #pragma unroll
  for (int t = 0; t < 4; ++t) {
    const int cc = t * 16 + cl;
    if (cc < ncols) {
      const int cg = col0 + cc;
      #pragma unroll
      for (int i = 0; i < 8; ++i) {
        const int n = ty * 16 + rb + i;
        const long o = (long)n * V3 + cg;
        v_sp[o] = acc[t][i] + v_sp[o];   // read-once/write-once, safe in place
      }
    }
  }
}

// K5: kinematic chain -> relative transforms A (N x 35 x 4x4). One lane per
// body; per-lane chain state lives in LDS (only feasible with 320KB/WGP LDS).
__global__ void k_chain(const float* __restrict__ Rs,
                        const float* __restrict__ Jp,
                        const float* __restrict__ scales,
                        float* __restrict__ Amat) {
  __shared__ float ch[32 * NJ * 12];
  const int tid = threadIdx.x;           // 0..31
  const int n = blockIdx.x * 32 + tid;
  float* C = ch + tid * (NJ * 12);
  const float* Rn = Rs + (long)n * (NJ * 9);
  const float* Jn = Jp + (long)n * (NJ * 3);
  const float* Sn = scales + (long)n * (NJ * 3);

  #pragma unroll
  for (int a = 0; a < 3; ++a) {
    C[a * 4 + 0] = Rn[a * 3 + 0];
    C[a * 4 + 1] = Rn[a * 3 + 1];
    C[a * 4 + 2] = Rn[a * 3 + 2];
    C[a * 4 + 3] = Jn[a];
  }
  for (int j = 1; j < NJ; ++j) {
    const int p = c_parents[j];
    float loc[12];
    #pragma unroll
    for (int a = 0; a < 3; ++a) {
      const float inv = 1.0f / Sn[p * 3 + a];
      loc[a * 4 + 0] = Rn[j * 9 + a * 3 + 0] * Sn[j * 3 + 0] * inv;
      loc[a * 4 + 1] = Rn[j * 9 + a * 3 + 1] * Sn[j * 3 + 1] * inv;
      loc[a * 4 + 2] = Rn[j * 9 + a * 3 + 2] * Sn[j * 3 + 2] * inv;
      loc[a * 4 + 3] = Jn[j * 3 + a] - Jn[p * 3 + a];
    }
    const float* P = C + p * 12;
    float* O = C + j * 12;
    #pragma unroll
    for (int a = 0; a < 3; ++a) {
      #pragma unroll
      for (int cc = 0; cc < 4; ++cc) {
        float acc = (cc == 3) ? P[a * 4 + 3] : 0.0f;
        acc += P[a * 4 + 0] * loc[0 * 4 + cc];
        acc += P[a * 4 + 1] * loc[1 * 4 + cc];
        acc += P[a * 4 + 2] * loc[2 * 4 + cc];
        O[a * 4 + cc] = acc;
      }
    }
  }
  float* Ao = Amat + (long)n * (NJ * 16);
  for (int j = 0; j < NJ; ++j) {
    const float* G = C + j * 12;
    const float jx = Jn[j * 3 + 0], jy = Jn[j * 3 + 1], jz = Jn[j * 3 + 2];
    #pragma unroll
    for (int a = 0; a < 3; ++a) {
      Ao[j * 16 + a * 4 + 0] = G[a * 4 + 0];
      Ao[j * 16 + a * 4 + 1] = G[a * 4 + 1];
      Ao[j * 16 + a * 4 + 2] = G[a * 4 + 2];
      Ao[j * 16 + a * 4 + 3] =
          G[a * 4 + 3] - (G[a * 4] * jx + G[a * 4 + 1] * jy + G[a * 4 + 2] * jz);
    }
    Ao[j * 16 + 12] = 0.f; Ao[j * 16 + 13] = 0.f;
    Ao[j * 16 + 14] = 0.f; Ao[j * 16 + 15] = 1.f;
  }
}

// K6: skinning. TDM stages the per-batch joint-transform panel A[n] (35x16
// f32, 2240B) into LDS once per workgroup (TENSORcnt), then each wave runs the
// WMMA GEMM T = weights(16x35) @ A[n](35x16) with B fed from DS, fused through
// LDS with the homogeneous apply + trans.
__global__ void k_skin(const float* __restrict__ weights,
                       const float* __restrict__ Amat,
                       const float* __restrict__ v_posed,
                       const float* __restrict__ trans,
                       float* __restrict__ verts) {
  __shared__ float Ab[NJ * 16];          // staged A[n] panel (560 f32)
  __shared__ float Tlds[4][16][16];
  const int wid = threadIdx.x >> 5, lane = threadIdx.x & 31;
  const int vt = blockIdx.x * 4 + wid;   // vertex tile (244 tiles, exact fit)
  const int n = blockIdx.y;
  const int v0 = vt * 16;
  const int mvalid = (NV - v0 < 16) ? (NV - v0) : 16;

#if __has_builtin(__builtin_amdgcn_tensor_load_to_lds) && \
    __has_builtin(__builtin_amdgcn_s_wait_tensorcnt)
  if (threadIdx.x < 32) {                // wave 0 drives the TDM
    const unsigned long long ga =
        (unsigned long long)(const void*)(Amat + (long)n * (NJ * 16));
    const unsigned lds_addr =
        (unsigned)(unsigned long long)(const void*)&Ab[0];
    // D# group 0: count=1 | lds_addr | global_addr[56:0] | type=2
    v4u g0;
    g0.x = 1u;
    g0.y = lds_addr;
    g0.z = (unsigned)(ga & 0xFFFFFFFFu);
    g0.w = (unsigned)((ga >> 32) & 0x01FFFFFFu) | (2u << 30);
    // D# group 1: data_size=4B; tensor_dim0=560, tensor_dim1=1;
    // tile_dim0=560, tile_dim1=1; tensor_dim0_stride=560.
    v8i g1;
    g1[0] = (int)(2u << 16);
    g1[1] = (int)((560u & 0xFFFFu) << 16);
    g1[2] = (int)((560u >> 16) | (1u << 16));
    g1[3] = (int)(560u << 16);
    g1[4] = 1;
    g1[5] = 560;
    g1[6] = 0;
    g1[7] = 0;
    const v4i z4 = {0, 0, 0, 0};
    const v8i z8 = {0, 0, 0, 0, 0, 0, 0, 0};
    __builtin_amdgcn_tensor_load_to_lds(g0, g1, z4, z4, z8, 0);
    __builtin_amdgcn_s_wait_tensorcnt(0);
  }
  __syncthreads();
#else
  for (int i = threadIdx.x; i < NJ * 16; i += blockDim.x)
    Ab[i] = Amat[(long)n * (NJ * 16) + i];
  __syncthreads();
#endif

  v8f acc = wmma_tile_f32(weights + (long)v0 * NJ, NJ, 1, mvalid,
                          Ab, 16, 16, NJ);
  {
    const int col = lane & 15, rb = (lane >> 4) << 3;
    #pragma unroll
    for (int i = 0; i < 8; ++i) Tlds[wid][rb + i][col] = acc[i];
  }
  __syncthreads();
  if (lane < 16 && lane < mvalid) {
    const int v = v0 + lane;
    const long vo = (long)n * V3 + (long)v * 3;
    const float x = v_posed[vo], y = v_posed[vo + 1], z = v_posed[vo + 2];
    const float* T = &Tlds[wid][lane][0];
    #pragma unroll
    for (int a = 0; a < 3; ++a)
      verts[vo + a] = T[a * 4 + 0] * x + T[a * 4 + 1] * y + T[a * 4 + 2] * z +
                      T[a * 4 + 3] + trans[n * 3 + a];
  }
}

extern "C" void kernel_launch(void* const* d_in, const int* in_sizes, int n_in,
                              void* d_out, int out_size, void* d_ws, size_t ws_size,
                              hipStream_t stream) {
  const float* beta        = (const float*)d_in[0];
  const float* theta       = (const float*)d_in[1];
  const float* betas_extra = (const float*)d_in[2];
  const float* deform      = (const float*)d_in[3];
  const float* trans       = (const float*)d_in[4];
  const float* v_template  = (const float*)d_in[5];
  const float* shapedirs   = (const float*)d_in[6];
  const float* posedirs    = (const float*)d_in[7];
  const float* Jreg        = (const float*)d_in[8];
  const float* weights     = (const float*)d_in[9];

  float* out    = (float*)d_out;
  float* verts  = out;                                   // 512*11667
  float* joints = verts + (size_t)NB * V3;               // 512*105
  float* Rs     = joints + (size_t)NB * NJ * 3;          // 512*315

  float* ws       = (float*)d_ws;
  float* v_shaped = ws;                                  // 512*11667 (reused as v_posed)
  float* Jp       = v_shaped + (size_t)NB * V3;          // 512*105
  float* pf       = Jp + (size_t)NB * NJ * 3;            // 512*306
  float* Amat     = pf + (size_t)NB * NPF;               // 512*560

  k_shape_blend<<<dim3(183, 32), 32, 0, stream>>>(beta, shapedirs, v_template,
                                                  deform, v_shaped);
  k_jreg<<<dim3(3, 32, 3), 32, 0, stream>>>(v_shaped, Jreg, Jp);
  k_rodrigues<<<dim3((NB * NJ + 127) / 128), 128, 0, stream>>>(theta, Rs, pf);
  k_pose_blend<<<dim3(183, 32), 32, 0, stream>>>(pf, posedirs, v_shaped);
  k_chain<<<dim3(16), 32, 0, stream>>>(Rs, Jp, betas_extra, Amat);
  k_skin<<<dim3(61, 512), 128, 0, stream>>>(weights, Amat, v_shaped, trans, verts);
  k_jreg<<<dim3(3, 32, 3), 32, 0, stream>>>(verts, Jreg, joints);
}